// GraphTransformerLayer_65360812310778
// MI455X (gfx1250) — compile-verified
//
#include <hip/hip_runtime.h>
#include <hip/hip_bf16.h>

// ---------------------------------------------------------------------------
// Graph transformer layer, fp32 end-to-end, GEMMs via V_WMMA_F32_16X16X4_F32.
// N = 50000 (= 3125 * 16, no tail), IN/OUT = 128, H = 8, D = 16, E = 1.6M.
// Edge stage (the roofline bottleneck: ~2.4 GB of L2-resident gathers) uses
// one wave32 per edge with float4 (global_load_b128) gathers.
// ---------------------------------------------------------------------------

typedef __attribute__((ext_vector_type(2))) float v2f;
typedef __attribute__((ext_vector_type(8))) float v8f;

#define NNODE   50000
#define NCH     128
#define EPSBN   1e-5f

// workspace layout (float offsets)
#define OQ    ((size_t)0)          // Q  [N*128]  (later reused as h1)
#define OK_   ((size_t)6400000)    // K  [N*128]  (later t lower half)
#define OV    ((size_t)12800000)   // V  [N*128]  (later t upper half; K..V contiguous = t[N*256])
#define OWV   ((size_t)19200000)   // wV [N*128]  (later h_att, later h2)
#define OZ    ((size_t)25600000)   // z  [N*8]
#define OSUM  ((size_t)26000000)   // sum[128], sumsq[128]

// ---------------------------------------------------------------------------
// Generic WMMA fp32 GEMM: out[M x ncols] = op( X[M x KDIM] @ W[KDIM x ncols] + bias )
// One wave -> one 16x16 tile. block = 256 threads = 8 waves (8 column tiles).
// grid.x = row tiles, grid.y = column-tile groups of 8.
// ---------------------------------------------------------------------------
template <int KDIM, bool RELU, bool RESID>
__global__ void gemm_wmma(const float* __restrict__ X, const float* __restrict__ W,
                          const float* __restrict__ bias, const float* __restrict__ resid,
                          float* __restrict__ out, int ncols) {
    const int lane = threadIdx.x & 31;
    const int wave = threadIdx.x >> 5;
    const int l16  = lane & 15;
    const int half = lane >> 4;                    // 0 or 1
    const int row0 = blockIdx.x * 16;
    const int col0 = (blockIdx.y * 8 + wave) * 16;

    // A (16x4 f32): lanes 0-15 -> M=l16, K = kb+0/kb+1 ; lanes 16-31 -> K = kb+2/kb+3
    const float* arow = X + (size_t)(row0 + l16) * KDIM + 2 * half;
    // B (4x16 f32): VGPR0 = B[kb + 2*half][col0+l16], VGPR1 = B[kb+1 + 2*half][...]
    const float* wb = W + (col0 + l16) + (size_t)(2 * half) * ncols;

    v8f acc = {0.f, 0.f, 0.f, 0.f, 0.f, 0.f, 0.f, 0.f};
#pragma unroll 8
    for (int kb = 0; kb < KDIM; kb += 4) {
        v2f a, b;
        a[0] = arow[kb];
        a[1] = arow[kb + 1];
        b[0] = wb[(size_t)kb * ncols];
        b[1] = wb[(size_t)(kb + 1) * ncols];
        acc = __builtin_amdgcn_wmma_f32_16x16x4_f32(false, a, false, b,
                                                    (short)0, acc, false, false);
    }

    const float bv = bias[col0 + l16];
#pragma unroll
    for (int r = 0; r < 8; ++r) {
        const int row = row0 + r + 8 * half;       // C/D layout: VGPR r -> M = r + 8*half
        float v = acc[r] + bv;
        if (RELU)  v = fmaxf(v, 0.0f);
        if (RESID) v += resid[(size_t)row * ncols + col0 + l16];
        out[(size_t)row * ncols + col0 + l16] = v;
    }
}

// ---------------------------------------------------------------------------
// Zero a float region (accumulators + stats)
// ---------------------------------------------------------------------------
__global__ void zero_kernel(float* __restrict__ p, size_t n) {
    size_t i = (size_t)blockIdx.x * blockDim.x + threadIdx.x;
    if (i < n) p[i] = 0.0f;
}

// ---------------------------------------------------------------------------
// Edge attention: one wave32 per edge, float4 (b128) gathers.
// Lane l owns channels c = 4l .. 4l+3, which all lie in head (l >> 2).
// Each head's 16 channels live in lanes 4h..4h+3, so a 2-step shfl_xor
// (offsets 1, 2) completes the per-head dot product. Scatter via f32 atomics.
// ---------------------------------------------------------------------------
__global__ void edge_attn(const float* __restrict__ Q, const float* __restrict__ Km,
                          const float* __restrict__ Vm, const int* __restrict__ src,
                          const int* __restrict__ dst, float* __restrict__ wV,
                          float* __restrict__ z, int nedge) {
    const int edge = (int)(((size_t)blockIdx.x * blockDim.x + threadIdx.x) >> 5);
    const int lane = threadIdx.x & 31;
    if (edge >= nedge) return;
    const int s = src[edge];
    const int d = dst[edge];

    const float4 q4 = ((const float4*)(Q  + (size_t)d * NCH))[lane];   // global_load_b128
    const float4 k4 = ((const float4*)(Km + (size_t)s * NCH))[lane];
    const float4 v4 = ((const float4*)(Vm + (size_t)s * NCH))[lane];

    // per-lane partial of this head's dot product
    float p = q4.x * k4.x + q4.y * k4.y + q4.z * k4.z + q4.w * k4.w;
    p += __shfl_xor(p, 1, 32);
    p += __shfl_xor(p, 2, 32);                     // 4-lane group = 16 channels of head (lane>>2)

    float t = p * 0.25f;                           // / sqrt(D), D = 16
    t = fminf(fmaxf(t, -5.0f), 5.0f);
    const float sc = __expf(t);

    float* wvd = wV + (size_t)d * NCH + 4 * lane;
    atomicAdd(&wvd[0], v4.x * sc);
    atomicAdd(&wvd[1], v4.y * sc);
    atomicAdd(&wvd[2], v4.z * sc);
    atomicAdd(&wvd[3], v4.w * sc);
    if ((lane & 3) == 0)
        atomicAdd(&z[(size_t)d * 8 + (lane >> 2)], sc);
}

// h_att = wV / z  (in place over wV), float4 per thread
__global__ void finalize_attn(float* __restrict__ wV, const float* __restrict__ z,
                              size_t total4) {
    size_t i = (size_t)blockIdx.x * blockDim.x + threadIdx.x;   // index in float4 units
    if (i >= total4) return;
    const size_t node = i >> 5;                    // / 32 float4s per row
    const int c0 = (int)((i & 31) << 2);           // base channel (multiple of 4, same head)
    const float invz = 1.0f / z[node * 8 + (c0 >> 4)];
    float4 v = ((float4*)wV)[i];
    v.x *= invz; v.y *= invz; v.z *= invz; v.w *= invz;
    ((float4*)wV)[i] = v;
}

// batch-norm stats: per-channel sum & sum-of-squares (atomics into sums[256])
__global__ void bn_stats(const float* __restrict__ h, float* __restrict__ sums, int n) {
    const int c = threadIdx.x;                     // 128 threads/block
    float s = 0.f, s2 = 0.f;
    for (int r = blockIdx.x; r < n; r += gridDim.x) {
        const float v = h[(size_t)r * NCH + c];
        s  += v;
        s2 += v * v;
    }
    atomicAdd(&sums[c], s);
    atomicAdd(&sums[NCH + c], s2);
}

__global__ void bn_apply(const float* __restrict__ h, const float* __restrict__ sums,
                         const float* __restrict__ gamma, const float* __restrict__ beta,
                         float* __restrict__ out, int n) {
    size_t i = (size_t)blockIdx.x * blockDim.x + threadIdx.x;   // float4 index
    if (i >= (size_t)n * (NCH / 4)) return;
    const int c0 = (int)((i & 31) << 2);
    const float invn = 1.0f / (float)n;
    const float4 hv = ((const float4*)h)[i];
    float4 o;
    {
        const float mu = sums[c0 + 0] * invn;
        const float vr = sums[NCH + c0 + 0] * invn - mu * mu;
        o.x = (hv.x - mu) * rsqrtf(vr + EPSBN) * gamma[c0 + 0] + beta[c0 + 0];
    }
    {
        const float mu = sums[c0 + 1] * invn;
        const float vr = sums[NCH + c0 + 1] * invn - mu * mu;
        o.y = (hv.y - mu) * rsqrtf(vr + EPSBN) * gamma[c0 + 1] + beta[c0 + 1];
    }
    {
        const float mu = sums[c0 + 2] * invn;
        const float vr = sums[NCH + c0 + 2] * invn - mu * mu;
        o.z = (hv.z - mu) * rsqrtf(vr + EPSBN) * gamma[c0 + 2] + beta[c0 + 2];
    }
    {
        const float mu = sums[c0 + 3] * invn;
        const float vr = sums[NCH + c0 + 3] * invn - mu * mu;
        o.w = (hv.w - mu) * rsqrtf(vr + EPSBN) * gamma[c0 + 3] + beta[c0 + 3];
    }
    ((float4*)out)[i] = o;
}

// ---------------------------------------------------------------------------
extern "C" void kernel_launch(void* const* d_in, const int* in_sizes, int n_in,
                              void* d_out, int out_size, void* d_ws, size_t ws_size,
                              hipStream_t stream) {
    const float* x_src = (const float*)d_in[0];
    const float* x_dst = (const float*)d_in[1];
    const int*   src   = (const int*)d_in[2];
    const int*   dst   = (const int*)d_in[3];
    const float* Wq = (const float*)d_in[4];  const float* bq = (const float*)d_in[5];
    const float* Wk = (const float*)d_in[6];  const float* bk = (const float*)d_in[7];
    const float* Wv = (const float*)d_in[8];  const float* bv = (const float*)d_in[9];
    const float* Wo = (const float*)d_in[10]; const float* bo = (const float*)d_in[11];
    const float* W1 = (const float*)d_in[12]; const float* b1 = (const float*)d_in[13];
    const float* W2 = (const float*)d_in[14]; const float* b2 = (const float*)d_in[15];
    const float* gamma2 = (const float*)d_in[16];
    const float* beta2  = (const float*)d_in[17];

    const int E_ = in_sizes[2];
    const int n  = in_sizes[0] / NCH;              // 50000
    const int rowTiles = n / 16;                   // 3125 (exact)

    float* ws  = (float*)d_ws;
    float* Qb  = ws + OQ;
    float* Kb  = ws + OK_;
    float* Vb  = ws + OV;
    float* wVb = ws + OWV;
    float* zb  = ws + OZ;
    float* sums = ws + OSUM;
    float* h1 = Qb;                                // reuse: Q dead after edge stage
    float* tb = Kb;                                // reuse: K..V contiguous -> t[N*256]
    float* h2 = wVb;                               // reuse: h_att dead after Wo GEMM

    // 1-3) Q = x_dst@Wq+bq ; K = x_src@Wk+bk ; V = x_src@Wv+bv
    gemm_wmma<128, false, false><<<dim3(rowTiles, 1), 256, 0, stream>>>(x_dst, Wq, bq, nullptr, Qb, NCH);
    gemm_wmma<128, false, false><<<dim3(rowTiles, 1), 256, 0, stream>>>(x_src, Wk, bk, nullptr, Kb, NCH);
    gemm_wmma<128, false, false><<<dim3(rowTiles, 1), 256, 0, stream>>>(x_src, Wv, bv, nullptr, Vb, NCH);

    // 4) zero wV, z, stats (contiguous region starting at OWV)
    {
        const size_t nz = (OSUM + 2 * NCH) - OWV;
        zero_kernel<<<(unsigned)((nz + 255) / 256), 256, 0, stream>>>(wVb, nz);
    }

    // 5) edge attention scatter (one wave32 per edge)
    {
        const size_t threads = (size_t)E_ * 32;
        edge_attn<<<(unsigned)((threads + 255) / 256), 256, 0, stream>>>(Qb, Kb, Vb, src, dst, wVb, zb, E_);
    }

    // 6) h_att = wV / z
    {
        const size_t total4 = (size_t)n * (NCH / 4);
        finalize_attn<<<(unsigned)((total4 + 255) / 256), 256, 0, stream>>>(wVb, zb, total4);
    }

    // 7) h1 = x_dst + h_att @ Wo + bo
    gemm_wmma<128, false, true><<<dim3(rowTiles, 1), 256, 0, stream>>>(wVb, Wo, bo, x_dst, h1, NCH);

    // 8) t = relu(h1 @ W1 + b1)   [N x 256]
    gemm_wmma<128, true, false><<<dim3(rowTiles, 2), 256, 0, stream>>>(h1, W1, b1, nullptr, tb, 2 * NCH);

    // 9) h2 = h1 + t @ W2 + b2    [N x 128]
    gemm_wmma<256, false, true><<<dim3(rowTiles, 1), 256, 0, stream>>>(tb, W2, b2, h1, h2, NCH);

    // 10) batch-norm stats + apply
    bn_stats<<<512, 128, 0, stream>>>(h2, sums, n);
    {
        const size_t total4 = (size_t)n * (NCH / 4);
        bn_apply<<<(unsigned)((total4 + 255) / 256), 256, 0, stream>>>(h2, sums, gamma2, beta2,
                                                                       (float*)d_out, n);
    }
}